// DCPDehazeGenerator_86397562126622
// MI455X (gfx1250) — compile-verified
//
#include <hip/hip_runtime.h>
#include <hip/hip_bf16.h>
#include <stdint.h>

// ---------------- geometry ----------------
#define Bn   16
#define Hn   512
#define Wn   512
#define HW   (Hn * Wn)          // 262144
#define PADn 16
#define PWn  544
#define PHn  544
#define FS   (PHn * PWn)        // 295936 floats per padded batch field
#define NF   ((size_t)Bn * FS)  // floats per workspace slot

// box / min-pool tiling
#define TW   64
#define TH   32
// min-pool (r=7) raw tile
#define MRH  46
#define MRW  78
// box (r=15) raw tile
#define RH2  62
#define RW2  94
#define HBS  72                 // hb row stride: 2*72 mod 64 = 16 -> conflict-free half-waves

typedef float    v2f  __attribute__((ext_vector_type(2)));
typedef float    v8f  __attribute__((ext_vector_type(8)));
typedef unsigned v4u  __attribute__((ext_vector_type(4)));
typedef int      v8i  __attribute__((ext_vector_type(8)));
typedef int      v4i  __attribute__((ext_vector_type(4)));

// ---------------- TDM tile load (global 2D tile -> LDS) ----------------
// Builds a D# (group0 + group1, groups 2/3 zero => 2D tensor) and issues
// tensor_load_to_lds (6-arg toolchain variant). Tile is tileW x tileH f32
// elements from a padded field with row stride PWn; all in-bounds.
__device__ __forceinline__ void tdm_load_tile(const float* gsrc, unsigned lds_off,
                                              int tileW, int tileH) {
  unsigned long long ga = (unsigned long long)(uintptr_t)gsrc;
  v4u g0;
  g0[0] = 1u;                                        // count = 1 (valid descriptor)
  g0[1] = lds_off;                                   // lds_addr (bytes)
  g0[2] = (unsigned)(ga & 0xffffffffu);              // global_addr[31:0]
  g0[3] = (unsigned)((ga >> 32) & 0x1ffffffu) | (2u << 30); // addr[56:32] | type=2
  v8i g1;
  g1[0] = (int)(2u << 16);                           // data_size = 2 (4 bytes)
  g1[1] = (int)(((unsigned)PWn & 0xffffu) << 16);    // tensor_dim0[15:0]
  g1[2] = (int)(((unsigned)PHn & 0xffffu) << 16);    // dim0[31:16]=0 | tensor_dim1[15:0]
  g1[3] = (int)(((unsigned)tileW & 0xffffu) << 16);  // dim1[31:16]=0 | tile_dim0
  g1[4] = (int)((unsigned)tileH & 0xffffu);          // tile_dim1 | tile_dim2=0
  g1[5] = (int)PWn;                                  // tensor_dim0_stride[31:0]
  g1[6] = (int)(((unsigned)FS & 0xffffu) << 16);     // stride0[47:32]=0 | stride1[15:0]
  g1[7] = (int)((unsigned)FS >> 16);                 // stride1[47:16]
  v4i gz4 = {0, 0, 0, 0};
  v8i gz8 = {0, 0, 0, 0, 0, 0, 0, 0};
  __builtin_amdgcn_tensor_load_to_lds(g0, g1, gz4, gz4, gz8, 0);
}

__device__ __forceinline__ int idxP(int b, int y, int x) {
  return b * FS + (y + PADn) * PWn + (x + PADn);
}

// ---------------- K1: channel-min + guidance (padded, with pad values) ----------------
__global__ __launch_bounds__(256) void prep_kernel(const float* __restrict__ x,
                                                   float* __restrict__ m,
                                                   float* __restrict__ g) {
  int i = blockIdx.x * blockDim.x + threadIdx.x;
  if (i >= Bn * PHn * PWn) return;
  int px = i % PWn; int t = i / PWn; int py = t % PHn; int b = t / PHn;
  int o = b * FS + py * PWn + px;
  if (py >= PADn && py < PADn + Hn && px >= PADn && px < PADn + Wn) {
    int y = py - PADn, xx = px - PADn;
    const float* xb = x + (size_t)b * 3 * HW + (size_t)y * Wn + xx;
    float r = xb[0], gg = xb[HW], bl = xb[2 * HW];
    float ip0 = (r + 1.f) * 0.5f, ip1 = (gg + 1.f) * 0.5f, ip2 = (bl + 1.f) * 0.5f;
    m[o] = fminf(ip0, fminf(ip1, ip2));
    g[o] = (0.2989f * r + 0.587f * gg + 0.114f * bl + 1.f) * 0.5f;   // (luma+1)/2
  } else {
    m[o] = 1.0f;   // min-pool pad value
    g[o] = 0.0f;   // box-sum pad value
  }
}

// ---------------- 15x15 separable min-pool (pad already materialized) ----------------
__global__ __launch_bounds__(256) void minpool_kernel(const float* __restrict__ src,
                                                      float* __restrict__ dst,
                                                      float alpha, float beta) {
  __shared__ float raw[MRH * MRW];
  __shared__ float hm[MRH * 64];
  int b = blockIdx.z, yt = blockIdx.y * TH, xt = blockIdx.x * TW;
  const float* sb = src + b * FS + (yt - 7 + PADn) * PWn + (xt - 7 + PADn);
#pragma unroll 1
  for (int e = threadIdx.x; e < MRH * MRW; e += 256) {
    int r = e / MRW, c = e - r * MRW;
    raw[e] = sb[r * PWn + c];
  }
  __syncthreads();
#pragma unroll 1
  for (int e = threadIdx.x; e < MRH * 64; e += 256) {
    int r = e >> 6, c = e & 63;
    float v = raw[r * MRW + c];
#pragma unroll
    for (int w = 1; w < 15; ++w) v = fminf(v, raw[r * MRW + c + w]);
    hm[e] = v;
  }
  __syncthreads();
  int xx = threadIdx.x & 63, yb = threadIdx.x >> 6;
#pragma unroll 1
  for (int k = 0; k < 8; ++k) {
    int y = yb * 8 + k;
    float v = hm[y * 64 + xx];
#pragma unroll
    for (int w = 1; w < 15; ++w) v = fminf(v, hm[(y + w) * 64 + xx]);
    dst[idxP(b, yt + y, xt + xx)] = beta + alpha * v;
  }
}

// ---------------- atmospheric light: histogram threshold + argmax intensity ----------------
__global__ __launch_bounds__(1024) void aselect_kernel(const float* __restrict__ dc,
                                                       const float* __restrict__ x,
                                                       float* __restrict__ A) {
  __shared__ unsigned hist[1024];
  __shared__ float sv[1024];
  __shared__ int   si[1024];
  __shared__ float sthr;
  int b = blockIdx.x, tid = threadIdx.x;
  hist[tid] = 0u;
  __syncthreads();
  const float* dcb = dc + b * FS;
#pragma unroll 1
  for (int i = tid; i < HW; i += 1024) {
    int y = i >> 9, xx = i & 511;
    float v = dcb[(y + PADn) * PWn + xx + PADn];
    int bin = (int)(v * 1024.f);
    bin = bin < 0 ? 0 : (bin > 1023 ? 1023 : bin);
    atomicAdd(&hist[bin], 1u);
  }
  __syncthreads();
  if (tid == 0) {
    unsigned cum = 0; int t = 1023;
    for (; t > 0; --t) { cum += hist[t]; if (cum >= 262u) break; }   // k = int(HW*0.001)
    sthr = (float)t * (1.f / 1024.f);
  }
  __syncthreads();
  float thr = sthr;
  const float* xb = x + (size_t)b * 3 * HW;
  float best = -1e30f; int bi = 0x7fffffff;
#pragma unroll 1
  for (int i = tid; i < HW; i += 1024) {
    int y = i >> 9, xx = i & 511;
    float v = dcb[(y + PADn) * PWn + xx + PADn];
    if (v >= thr) {
      float r = xb[i], gg = xb[i + HW], bl = xb[i + 2 * HW];
      float inten = 0.2989f * (r + 1.f) * 0.5f + 0.587f * (gg + 1.f) * 0.5f +
                    0.114f * (bl + 1.f) * 0.5f;
      if (inten > best || (inten == best && i < bi)) { best = inten; bi = i; }
    }
  }
  sv[tid] = best; si[tid] = bi;
  __syncthreads();
  for (int s = 512; s > 0; s >>= 1) {
    if (tid < s) {
      float ov = sv[tid + s]; int oi = si[tid + s];
      if (ov > sv[tid] || (ov == sv[tid] && oi < si[tid])) { sv[tid] = ov; si[tid] = oi; }
    }
    __syncthreads();
  }
  if (tid == 0) {
    int i = si[0];
    A[b * 3 + 0] = (xb[i] + 1.f) * 0.5f;
    A[b * 3 + 1] = (xb[i + HW] + 1.f) * 0.5f;
    A[b * 3 + 2] = (xb[i + 2 * HW] + 1.f) * 0.5f;
  }
}

// ---------------- channel-min of imgPatch / A_patch ----------------
__global__ __launch_bounds__(256) void m2_kernel(const float* __restrict__ x,
                                                 const float* __restrict__ A,
                                                 float* __restrict__ m2) {
  int i = blockIdx.x * blockDim.x + threadIdx.x;
  if (i >= Bn * PHn * PWn) return;
  int px = i % PWn; int t = i / PWn; int py = t % PHn; int b = t / PHn;
  int o = b * FS + py * PWn + px;
  if (py >= PADn && py < PADn + Hn && px >= PADn && px < PADn + Wn) {
    int y = py - PADn, xx = px - PADn;
    size_t pix = (size_t)y * Wn + xx;
    const float* xb = x + (size_t)b * 3 * HW;
    float v0 = (xb[pix] + 1.f) * 0.5f / A[b * 3 + 0];
    float v1 = (xb[pix + HW] + 1.f) * 0.5f / A[b * 3 + 1];
    float v2 = (xb[pix + 2 * HW] + 1.f) * 0.5f / A[b * 3 + 2];
    m2[o] = fminf(v0, fminf(v1, v2));
  } else {
    m2[o] = 1.0f;
  }
}

__global__ __launch_bounds__(256) void fill_border_kernel(float* __restrict__ p, float val) {
  int i = blockIdx.x * blockDim.x + threadIdx.x;
  if (i >= Bn * PHn * PWn) return;
  int px = i % PWn; int t = i / PWn; int py = t % PHn; int b = t / PHn;
  if (py < PADn || py >= PADn + Hn || px < PADn || px >= PADn + Wn)
    p[b * FS + py * PWn + px] = val;
}

// Vertical 31-tap box sum via WMMA: D(16x16) += band(16x4) x hsum(4x16).
// band(m,k) = 1 iff 0 <= k-m <= 30  (one unsigned compare per element).
__device__ __forceinline__ v8f wmma_vbox_step(v8f acc, const float* hb, int kbase,
                                              int hi2, int mrow, int wy, int wx, int lane) {
  int k0 = kbase + hi2 * 2;
  int m = wy * 16 + mrow;
  v2f a, bf;
  a.x = ((unsigned)(k0 - m) <= 30u) ? 1.f : 0.f;
  a.y = ((unsigned)(k0 + 1 - m) <= 30u) ? 1.f : 0.f;
  int col = wx * 16 + (lane & 15);
  bf.x = hb[k0 * HBS + col];
  bf.y = hb[(k0 + 1) * HBS + col];
  return __builtin_amdgcn_wmma_f32_16x16x4_f32(false, a, false, bf, (short)0, acc,
                                               false, false);
}

// Sliding-window horizontal 31-tap sums: thread t owns row r = t>>2, 16 consecutive
// columns starting at (t&3)*16. 31 adds + 15 incremental updates per thread.
// f: 0 -> tX, 1 -> tY, 2 -> tX*tY, 3 -> tX*tX. Rows >= RH2 write zeros.
template <int F>
__device__ __forceinline__ void hsum_sliding(const float* tX, const float* tY,
                                             float* hb, int tid) {
  int r = tid >> 2;
  int cg = (tid & 3) * 16;
  int base = r * RW2 + cg;
  float* hrow = hb + r * HBS + cg;
  if (r < RH2) {
    float s = 0.f;
#pragma unroll
    for (int w = 0; w < 31; ++w) {
      int i = base + w;
      float v;
      if (F == 0)      v = tX[i];
      else if (F == 1) v = tY[i];
      else if (F == 2) v = tX[i] * tY[i];
      else             v = tX[i] * tX[i];
      s += v;
    }
    hrow[0] = s;
#pragma unroll
    for (int j = 1; j < 16; ++j) {
      int ia = base + 30 + j, is = base + j - 1;
      float va, vs;
      if (F == 0)      { va = tX[ia];          vs = tX[is]; }
      else if (F == 1) { va = tY[ia];          vs = tY[is]; }
      else if (F == 2) { va = tX[ia] * tY[ia]; vs = tX[is] * tY[is]; }
      else             { va = tX[ia] * tX[ia]; vs = tX[is] * tX[is]; }
      s += va - vs;
      hrow[j] = s;
    }
  } else {
#pragma unroll
    for (int j = 0; j < 16; ++j) hrow[j] = 0.f;
  }
}

// ---------------- guided filter stage 1: a,b from I(=g) and p ----------------
__global__ __launch_bounds__(256) void box1_kernel(const float* __restrict__ I,
                                                   const float* __restrict__ P,
                                                   float* __restrict__ aout,
                                                   float* __restrict__ bout) {
  __shared__ float tI[RH2 * RW2];
  __shared__ float tP[RH2 * RW2];
  __shared__ float hb[64 * HBS];      // rows 62,63 forced to 0
  int b = blockIdx.z, yt = blockIdx.y * TH, xt = blockIdx.x * TW;
  const float* Ib = I + b * FS + (yt + 1) * PWn + (xt + 1);   // (yt-15+PAD, xt-15+PAD)
  const float* Pb = P + b * FS + (yt + 1) * PWn + (xt + 1);
  if ((threadIdx.x >> 5) == 0) {
    tdm_load_tile(Ib, (unsigned)(uintptr_t)tI, RW2, RH2);
    tdm_load_tile(Pb, (unsigned)(uintptr_t)tP, RW2, RH2);
    __builtin_amdgcn_s_wait_tensorcnt(0);
  }
  __syncthreads();

  int tid = threadIdx.x;
  int lane = tid & 31;
  int wv = tid >> 5;                  // 8 waves -> 8 16x16 output subtiles
  int wy = wv >> 2, wx = wv & 3;
  int mrow = lane & 15;
  int hi2 = (lane >> 4) & 1;
  v8f acc[4] = {};

#pragma unroll
  for (int f = 0; f < 4; ++f) {
    if (f == 0)      hsum_sliding<0>(tI, tP, hb, tid);
    else if (f == 1) hsum_sliding<1>(tI, tP, hb, tid);
    else if (f == 2) hsum_sliding<2>(tI, tP, hb, tid);
    else             hsum_sliding<3>(tI, tP, hb, tid);
    __syncthreads();
#pragma unroll
    for (int kc = 0; kc < 16; ++kc)
      acc[f] = wmma_vbox_step(acc[f], hb, kc * 4, hi2, mrow, wy, wx, lane);
    __syncthreads();
  }

#pragma unroll
  for (int r = 0; r < 8; ++r) {
    int M = r + hi2 * 8;
    int y = yt + wy * 16 + M;
    int xx = xt + wx * 16 + (lane & 15);
    float Ny = fminf((float)(y + 15), 511.f) - fmaxf((float)(y - 15), 0.f) + 1.f;
    float Nx = fminf((float)(xx + 15), 511.f) - fmaxf((float)(xx - 15), 0.f) + 1.f;
    float Ninv = 1.f / (Ny * Nx);
    float mI = acc[0][r] * Ninv, mp = acc[1][r] * Ninv;
    float mIp = acc[2][r] * Ninv, mII = acc[3][r] * Ninv;
    float av = (mIp - mI * mp) / (mII - mI * mI + 1e-3f);
    float bv = mp - av * mI;
    int o = idxP(b, y, xx);
    aout[o] = av; bout[o] = bv;
  }
}

// ---------------- guided filter stage 2 + final compose ----------------
__global__ __launch_bounds__(256) void box2_kernel(const float* __restrict__ Af,
                                                   const float* __restrict__ Bf,
                                                   const float* __restrict__ G,
                                                   const float* __restrict__ x,
                                                   const float* __restrict__ Aatm,
                                                   float* __restrict__ out) {
  __shared__ float tA[RH2 * RW2];
  __shared__ float tB[RH2 * RW2];
  __shared__ float hb[64 * HBS];
  int b = blockIdx.z, yt = blockIdx.y * TH, xt = blockIdx.x * TW;
  const float* Ab = Af + b * FS + (yt + 1) * PWn + (xt + 1);
  const float* Bb = Bf + b * FS + (yt + 1) * PWn + (xt + 1);
  if ((threadIdx.x >> 5) == 0) {
    tdm_load_tile(Ab, (unsigned)(uintptr_t)tA, RW2, RH2);
    tdm_load_tile(Bb, (unsigned)(uintptr_t)tB, RW2, RH2);
    __builtin_amdgcn_s_wait_tensorcnt(0);
  }
  __syncthreads();

  int tid = threadIdx.x;
  int lane = tid & 31;
  int wv = tid >> 5;
  int wy = wv >> 2, wx = wv & 3;
  int mrow = lane & 15;
  int hi2 = (lane >> 4) & 1;
  v8f acc[2] = {};

#pragma unroll
  for (int f = 0; f < 2; ++f) {
    if (f == 0) hsum_sliding<0>(tA, tB, hb, tid);
    else        hsum_sliding<1>(tA, tB, hb, tid);
    __syncthreads();
#pragma unroll
    for (int kc = 0; kc < 16; ++kc)
      acc[f] = wmma_vbox_step(acc[f], hb, kc * 4, hi2, mrow, wy, wx, lane);
    __syncthreads();
  }

  float* J    = out;
  float* Tout = out + (size_t)Bn * 3 * HW;
  float* Mout = Tout + (size_t)Bn * HW;
#pragma unroll
  for (int r = 0; r < 8; ++r) {
    int M = r + hi2 * 8;
    int y = yt + wy * 16 + M;
    int xx = xt + wx * 16 + (lane & 15);
    float Ny = fminf((float)(y + 15), 511.f) - fmaxf((float)(y - 15), 0.f) + 1.f;
    float Nx = fminf((float)(xx + 15), 511.f) - fmaxf((float)(xx - 15), 0.f) + 1.f;
    float Ninv = 1.f / (Ny * Nx);
    float ma = acc[0][r] * Ninv, mb = acc[1][r] * Ninv;
    float gv = G[idxP(b, y, xx)];
    float T = ma * gv + mb;
    size_t pix = (size_t)y * Wn + xx;
    Tout[(size_t)b * HW + pix] = T;
#pragma unroll
    for (int c = 0; c < 3; ++c) {
      float Ac = Aatm[b * 3 + c];
      float mA = 2.f * Ac - 1.f;
      float ip = (x[((size_t)b * 3 + c) * HW + pix] + 1.f) * 0.5f;
      J[((size_t)b * 3 + c) * HW + pix] = (ip - mA) / T + mA;
      Mout[((size_t)b * 3 + c) * HW + pix] = mA;
    }
  }
}

// ---------------- launcher ----------------
extern "C" void kernel_launch(void* const* d_in, const int* in_sizes, int n_in,
                              void* d_out, int out_size, void* d_ws, size_t ws_size,
                              hipStream_t stream) {
  (void)in_sizes; (void)n_in; (void)out_size; (void)ws_size;
  const float* x = (const float*)d_in[0];
  float* ws = (float*)d_ws;
  float* S0 = ws;
  float* S1 = ws + NF;
  float* S2 = ws + 2 * NF;
  float* S3 = ws + 3 * NF;
  float* Aatm = ws + 4 * NF;        // [B,3]

  int npad = Bn * PHn * PWn;
  dim3 blk(256);
  dim3 gpad((npad + 255) / 256);
  dim3 gt(Wn / TW, Hn / TH, Bn);    // (8,16,16)

  prep_kernel<<<gpad, blk, 0, stream>>>(x, S0, S1);                    // S0=m, S1=g
  minpool_kernel<<<gt, blk, 0, stream>>>(S0, S2, 1.f, 0.f);            // S2=dark channel
  aselect_kernel<<<Bn, 1024, 0, stream>>>(S2, x, Aatm);                // A per batch
  m2_kernel<<<gpad, blk, 0, stream>>>(x, Aatm, S2);                    // S2=min_c(ip/A)
  fill_border_kernel<<<gpad, blk, 0, stream>>>(S0, 0.f);               // S0 border -> box pad
  minpool_kernel<<<gt, blk, 0, stream>>>(S2, S0, -0.95f, 1.f);         // S0=trans_raw
  fill_border_kernel<<<gpad, blk, 0, stream>>>(S2, 0.f);               // borders for a,b
  fill_border_kernel<<<gpad, blk, 0, stream>>>(S3, 0.f);
  box1_kernel<<<gt, blk, 0, stream>>>(S1, S0, S2, S3);                 // S2=a, S3=b
  box2_kernel<<<gt, blk, 0, stream>>>(S2, S3, S1, x, Aatm, (float*)d_out);
}